// Aggregator_31808527794714
// MI455X (gfx1250) — compile-verified
//
#include <hip/hip_runtime.h>
#include <cstdint>

typedef float v2f __attribute__((ext_vector_type(2)));
typedef float v8f __attribute__((ext_vector_type(8)));

#define NEG_SLOPE 0.01f

// ---------------------------------------------------------------------------
// Zero the accumulation buffers (out, denom, counts) at the start of each call
// ---------------------------------------------------------------------------
__global__ void zero_kernel(float* __restrict__ out, float* __restrict__ denom,
                            float* __restrict__ counts, int n_out, int n_seg) {
    int i = blockIdx.x * blockDim.x + threadIdx.x;
    if (i < n_out) out[i] = 0.0f;
    if (i < n_seg) { denom[i] = 0.0f; counts[i] = 0.0f; }
}

// ---------------------------------------------------------------------------
// transformed[r,n,k] = sum_d ego[n,d] * W[r,d,k]
// One wave computes a 16-row x 64-col output stripe for one relation r using
// V_WMMA_F32_16X16X4_F32.  K-loop of 16 steps (K=64), A fragment reused across
// the 4 column tiles (4 x v8f accumulators).
//
// f32 WMMA VGPR layouts (ISA 7.12.2):
//   A 16x4 : lanes 0-15  -> {v0=K0, v1=K1}; lanes 16-31 -> {v0=K2, v1=K3}
//   B 4x16 : v0 -> {K0 | K2}, v1 -> {K1 | K3}; col = lane%16
//   C 16x16: row = vgpr + 8*(lane>=16); col = lane%16
// ---------------------------------------------------------------------------
__global__ void __launch_bounds__(256)
relation_gemm_kernel(const float* __restrict__ ego, const float* __restrict__ W,
                     float* __restrict__ transformed, int N, int rowTiles, int R) {
    const int lane = threadIdx.x & 31;
    const int wid  = blockIdx.x * (blockDim.x >> 5) + (threadIdx.x >> 5);
    const int r    = wid / rowTiles;
    if (r >= R) return;                       // whole-wave early exit only
    const int m0   = (wid % rowTiles) << 4;
    const int half = lane >> 4;               // 0: lanes 0-15, 1: lanes 16-31
    const int ml   = lane & 15;

    int arowIdx = m0 + ml;                    // clamp keeps EXEC full on loads
    if (arowIdx > N - 1) arowIdx = N - 1;
    const float* __restrict__ arow = ego + (size_t)arowIdx * 64 + 2 * half;
    const float* __restrict__ Wr   = W + (size_t)r * 64 * 64;

    v8f acc0 = {}, acc1 = {}, acc2 = {}, acc3 = {};
#pragma unroll
    for (int kb = 0; kb < 16; ++kb) {
        const int k0 = kb * 4;
        v2f a;
        a.x = arow[k0];
        a.y = arow[k0 + 1];
        const float* __restrict__ brow = Wr + (size_t)(k0 + 2 * half) * 64 + ml;
        v2f b0, b1, b2, b3;
        b0.x = brow[0];       b0.y = brow[64];
        b1.x = brow[16];      b1.y = brow[64 + 16];
        b2.x = brow[32];      b2.y = brow[64 + 32];
        b3.x = brow[48];      b3.y = brow[64 + 48];
        acc0 = __builtin_amdgcn_wmma_f32_16x16x4_f32(false, a, false, b0, (short)0, acc0, false, false);
        acc1 = __builtin_amdgcn_wmma_f32_16x16x4_f32(false, a, false, b1, (short)0, acc1, false, false);
        acc2 = __builtin_amdgcn_wmma_f32_16x16x4_f32(false, a, false, b2, (short)0, acc2, false, false);
        acc3 = __builtin_amdgcn_wmma_f32_16x16x4_f32(false, a, false, b3, (short)0, acc3, false, false);
    }

    float* __restrict__ outbase = transformed + ((size_t)r * N + m0) * 64;
#pragma unroll
    for (int v = 0; v < 8; ++v) {
        const int row = v + 8 * half;
        if (m0 + row < N) {
            float* __restrict__ orow = outbase + (size_t)row * 64 + ml;
            orow[0]  = acc0[v];
            orow[16] = acc1[v];
            orow[32] = acc2[v];
            orow[48] = acc3[v];
        }
    }
}

// ---------------------------------------------------------------------------
// Per-edge score: s = leaky_relu(dot(transformed[type,head,:], ego[tail,:]))
// 16 lanes per edge, each lane loads one float4 chunk (256B coalesced rows),
// shfl-xor tree reduction within the 16-lane subgroup.
// ---------------------------------------------------------------------------
__global__ void __launch_bounds__(256)
score_kernel(const float* __restrict__ ego, const float* __restrict__ transformed,
             const int* __restrict__ head, const int* __restrict__ tail,
             const int* __restrict__ etype, float* __restrict__ exp_scores,
             float* __restrict__ denom, float* __restrict__ counts,
             int E, int N) {
    const int tid = blockIdx.x * blockDim.x + threadIdx.x;
    const int e   = tid >> 4;
    const int ml  = tid & 15;
    if (e >= E) return;
    const int h = head[e];
    const int t = tail[e];
    const int r = etype[e];

    const float4* __restrict__ trow = (const float4*)(transformed + ((size_t)r * N + h) * 64);
    const float4* __restrict__ erow = (const float4*)(ego + (size_t)t * 64);
    const float4 a = trow[ml];
    const float4 b = erow[ml];
    float p = a.x * b.x + a.y * b.y + a.z * b.z + a.w * b.w;
    p += __shfl_xor(p, 1);
    p += __shfl_xor(p, 2);
    p += __shfl_xor(p, 4);
    p += __shfl_xor(p, 8);
    if (ml == 0) {
        const float s  = (p > 0.0f) ? p : p * NEG_SLOPE;
        const float ex = __expf(s);
        exp_scores[e] = ex;
        atomicAdd(&denom[h], ex);
        atomicAdd(&counts[h], 1.0f);
    }
}

// ---------------------------------------------------------------------------
// out[head,k] += ego[tail,k] * (exp_scores[e] / denom[head]).  One thread per
// (edge, k): 64 consecutive threads read one ego row coalesced.
// ---------------------------------------------------------------------------
__global__ void __launch_bounds__(256)
aggregate_kernel(const float* __restrict__ ego, const int* __restrict__ head,
                 const int* __restrict__ tail, const float* __restrict__ exp_scores,
                 const float* __restrict__ denom, float* __restrict__ out,
                 long long total) {
    const long long gid = (long long)blockIdx.x * blockDim.x + threadIdx.x;
    if (gid >= total) return;
    const int e = (int)(gid >> 6);
    const int k = (int)(gid & 63);
    const int h = head[e];
    const float att = exp_scores[e] / denom[h];
    const float val = ego[(size_t)tail[e] * 64 + k] * att;
    atomicAdd(&out[(size_t)h * 64 + k], val);
}

// ---------------------------------------------------------------------------
// out /= max(counts, 1)   (scatter-mean denominator)
// ---------------------------------------------------------------------------
__global__ void __launch_bounds__(256)
finalize_kernel(float* __restrict__ out, const float* __restrict__ counts, int n_out) {
    int i = blockIdx.x * blockDim.x + threadIdx.x;
    if (i < n_out) {
        const float c = counts[i >> 6];
        out[i] = out[i] / fmaxf(c, 1.0f);
    }
}

// ---------------------------------------------------------------------------
extern "C" void kernel_launch(void* const* d_in, const int* in_sizes, int n_in,
                              void* d_out, int out_size, void* d_ws, size_t ws_size,
                              hipStream_t stream) {
    const float* ego   = (const float*)d_in[0];   // [N,64]
    const float* W     = (const float*)d_in[1];   // [R,64,64]
    const int*   eidx  = (const int*)d_in[2];     // [2,E]
    const int*   etype = (const int*)d_in[3];     // [E]

    const int N = in_sizes[0] / 64;
    const int R = in_sizes[1] / (64 * 64);
    const int E = in_sizes[3];
    const int* head = eidx;
    const int* tail = eidx + E;

    // workspace layout
    float* transformed = (float*)d_ws;                          // R*N*64 floats
    float* exp_scores  = transformed + (size_t)R * N * 64;      // E floats
    float* denom       = exp_scores + E;                        // N floats
    float* counts      = denom + N;                             // N floats
    float* out         = (float*)d_out;                         // [N,64]

    const int n_out = N * 64;

    // 1) zero accumulators
    zero_kernel<<<(n_out + 255) / 256, 256, 0, stream>>>(out, denom, counts, n_out, N);

    // 2) WMMA GEMM: transformed[r,n,k]
    {
        const int rowTiles = (N + 15) / 16;
        const int waves    = R * rowTiles;
        const int blocks   = (waves + 7) / 8;   // 8 waves per 256-thread block
        relation_gemm_kernel<<<blocks, 256, 0, stream>>>(ego, W, transformed, N, rowTiles, R);
    }

    // 3) per-edge scores + segment-softmax denominator + counts
    {
        const long long threads = (long long)E * 16;
        const int blocks = (int)((threads + 255) / 256);
        score_kernel<<<blocks, 256, 0, stream>>>(ego, transformed, head, tail, etype,
                                                 exp_scores, denom, counts, E, N);
    }

    // 4) weighted scatter-sum into out
    {
        const long long total = (long long)E * 64;
        const int blocks = (int)((total + 255) / 256);
        aggregate_kernel<<<blocks, 256, 0, stream>>>(ego, head, tail, exp_scores, denom,
                                                     out, total);
    }

    // 5) scatter-mean divide
    finalize_kernel<<<(n_out + 255) / 256, 256, 0, stream>>>(out, counts, n_out);
}